// FairData_81406810128825
// MI455X (gfx1250) — compile-verified
//
#include <hip/hip_runtime.h>
#include <math.h>

// ---------------------------------------------------------------------------
// FairData loss on MI455X (gfx1250, wave32).
// Memory-bound gather/dot workload; batched row-dots are computed as the
// diagonal of 16x16 WMMA tiles using V_WMMA_F32_16X16X4_F32 (full f32).
// ---------------------------------------------------------------------------

#define B_SZ      262144
#define LEN_NOISE 26214      // int(B * 0.1)
#define DIM       64
#define CBLOCKS   1024       // B / 256, compaction blocks
#define GRID2     2048       // B / 128 (256 thr = 8 waves x 16 rows)
#define GRID3     2048       // covers up to B rows per gender

typedef __attribute__((ext_vector_type(2))) float v2f;
typedef __attribute__((ext_vector_type(8))) float v8f;

__device__ __forceinline__ float2 ld2(const float* p) {
    return *(const float2*)p;
}
__device__ __forceinline__ float clip1(float x) {
    return fminf(1.0f, fmaxf(-1.0f, x));
}
// -log_sigmoid(x) = softplus(-x); this computes softplus(z) stably.
__device__ __forceinline__ float softplus_(float z) {
    return fmaxf(z, 0.0f) + log1pf(expf(-fabsf(z)));
}
__device__ __forceinline__ float waveSum(float v) {
#pragma unroll
    for (int off = 16; off > 0; off >>= 1)
        v += __shfl_xor(v, off, 32);
    return v;
}
// Extract diag[m] (m = lane & 15) from a 16x16 f32 WMMA accumulator.
// Layout: (M,N): M<8 -> VGPR M, lane N ; M>=8 -> VGPR M-8, lane 16+N.
__device__ __forceinline__ float wmmaDiag(v8f c, int m) {
    float d = 0.0f;
#pragma unroll
    for (int k = 0; k < 8; ++k) {
        float lo = __shfl(c[k], k, 32);        // diag[k]
        float hi = __shfl(c[k], 24 + k, 32);   // diag[8+k]
        d = (m == k)     ? lo : d;
        d = (m == 8 + k) ? hi : d;
    }
    return d;
}

// --------------------------- compaction ------------------------------------

__global__ void __launch_bounds__(256) k_count(const int* __restrict__ ub,
                                               const int* __restrict__ uf,
                                               int* __restrict__ blkCnt) {
    int b = blockIdx.x * 256 + threadIdx.x;
    int male = (uf[ub[b]] != 0) ? 1 : 0;
    unsigned mb = (unsigned)__ballot(male);
    __shared__ int s[8];
    if ((threadIdx.x & 31) == 0) s[threadIdx.x >> 5] = __popc(mb);
    __syncthreads();
    if (threadIdx.x == 0) {
        int t = 0;
#pragma unroll
        for (int i = 0; i < 8; ++i) t += s[i];
        blkCnt[blockIdx.x] = t;
    }
}

__global__ void __launch_bounds__(1024) k_scan(const int* __restrict__ cnt,
                                               int* __restrict__ off,
                                               int* __restrict__ meta) {
    __shared__ int s[1024];
    int t = threadIdx.x;
    int c = cnt[t];
    s[t] = c;
    __syncthreads();
    for (int o = 1; o < 1024; o <<= 1) {
        int v = (t >= o) ? s[t - o] : 0;
        __syncthreads();
        s[t] += v;
        __syncthreads();
    }
    off[t] = s[t] - c;                 // exclusive prefix of male counts
    if (t == 1023) {
        meta[0] = s[1023];             // Nm
        meta[1] = B_SZ - s[1023];      // Nf
    }
}

__global__ void __launch_bounds__(256) k_scatter(const int* __restrict__ ub,
                                                 const int* __restrict__ uf,
                                                 const int* __restrict__ blkOff,
                                                 const int* __restrict__ meta,
                                                 int* __restrict__ list) {
    int t = threadIdx.x, blk = blockIdx.x;
    int b = blk * 256 + t;
    int male = (uf[ub[b]] != 0) ? 1 : 0;
    unsigned mb = (unsigned)__ballot(male);
    int lane = t & 31, wave = t >> 5;
    int rankW = __popc(mb & ((1u << lane) - 1u));
    __shared__ int wc[8];
    if (lane == 0) wc[wave] = __popc(mb);
    __syncthreads();
    int pre = 0;
    for (int i = 0; i < wave; ++i) pre += wc[i];
    int mRank = pre + rankW;                       // males strictly before t in block
    int Nm = meta[0];
    if (male)
        list[blkOff[blk] + mRank] = b;             // males at [0, Nm)
    else
        list[Nm + blk * 256 - blkOff[blk] + (t - mRank)] = b;  // females at [Nm, B)
}

// --------------------------- main loss pass --------------------------------

__global__ void __launch_bounds__(256) k_main(const int* __restrict__ ub,
                                              const int* __restrict__ ibv,
                                              const int* __restrict__ jbv,
                                              const float* __restrict__ uw,
                                              const float* __restrict__ iw,
                                              const float* __restrict__ nw,
                                              float* __restrict__ p2) {
    const int lane   = threadIdx.x & 31;
    const int wave   = threadIdx.x >> 5;
    const int m      = lane & 15;
    const int colOff = (lane < 16) ? 0 : 2;
    const int row    = (blockIdx.x * 8 + wave) * 16 + m;   // B is a multiple of 128

    const float* uRow = uw + (size_t)ub[row] * DIM;
    int  bi       = ibv[row];
    bool addNoise = (row >= (B_SZ - LEN_NOISE));
    const float* aI = iw + (size_t)bi * DIM;
    const float* aN = nw + (size_t)bi * DIM;
    bool jNoise = (row < LEN_NOISE);
    int  jb2    = jNoise ? jbv[B_SZ - LEN_NOISE + row] : jbv[row - LEN_NOISE];
    const float* ajI = iw + (size_t)jb2 * DIM;
    const float* ajN = nw + (size_t)jb2 * DIM;
    const float* jRow = iw + (size_t)jbv[row] * DIM;

    v8f cAdd = {};
    v8f cNeg = {};
    float sq = 0.0f;

#pragma unroll
    for (int c = 0; c < 16; ++c) {
        int col = 4 * c + colOff;
        float2 u2 = ld2(uRow + col);
        float2 a2 = ld2(aI + col);
        if (addNoise) {
            float2 n2 = ld2(aN + col);
            a2.x = clip1(a2.x) + n2.x;
            a2.y = clip1(a2.y) + n2.y;
        }
        float2 aj2 = ld2(ajI + col);
        if (jNoise) {
            float2 n2 = ld2(ajN + col);
            aj2.x = clip1(aj2.x) + n2.x;
            aj2.y = clip1(aj2.y) + n2.y;
        }
        float2 j2 = ld2(jRow + col);
        sq += u2.x * u2.x + u2.y * u2.y + a2.x * a2.x + a2.y * a2.y +
              j2.x * j2.x + j2.y * j2.y;
        v2f av;  av[0] = u2.x;  av[1] = u2.y;
        v2f bA;  bA[0] = a2.x;  bA[1] = a2.y;
        v2f bN;  bN[0] = aj2.x; bN[1] = aj2.y;
        cAdd = __builtin_amdgcn_wmma_f32_16x16x4_f32(false, av, false, bA,
                                                     (short)0, cAdd, false, false);
        cNeg = __builtin_amdgcn_wmma_f32_16x16x4_f32(false, av, false, bN,
                                                     (short)0, cNeg, false, false);
    }

    float predAdd = wmmaDiag(cAdd, m);
    float predNeg = wmmaDiag(cNeg, m);
    float lossC = 0.0f;
    if (lane < 16) lossC = softplus_(predNeg - predAdd);  // -log_sigmoid(add-neg)

    lossC = waveSum(lossC);
    float sqW = waveSum(sq);

    __shared__ float sl[8], ss[8];
    if (lane == 0) { sl[wave] = lossC; ss[wave] = sqW; }
    __syncthreads();
    if (threadIdx.x == 0) {
        float L = 0.0f, S = 0.0f;
#pragma unroll
        for (int i = 0; i < 8; ++i) { L += sl[i]; S += ss[i]; }
        p2[(size_t)blockIdx.x * 2 + 0] = L;
        p2[(size_t)blockIdx.x * 2 + 1] = S;
    }
}

// --------------------------- fake-positive pass ----------------------------

__global__ void __launch_bounds__(256) k_fake(const int* __restrict__ ub,
                                              const int* __restrict__ ibv,
                                              const int* __restrict__ jbv,
                                              const float* __restrict__ uw,
                                              const float* __restrict__ iw,
                                              const float* __restrict__ nw,
                                              const int* __restrict__ meta,
                                              const int* __restrict__ list,
                                              float* __restrict__ p3) {
    const int g  = blockIdx.y;                 // 0 = male, 1 = female
    const int Nm = meta[0], Nf = meta[1];
    const int N        = (g == 0) ? Nm : Nf;
    const int Nopp     = (g == 0) ? Nf : Nm;
    const int baseSelf = (g == 0) ? 0 : Nm;
    const int baseOpp  = (g == 0) ? Nm : 0;
    const int mod      = (Nopp > 1) ? (Nopp - 1) : 1;

    const int lane   = threadIdx.x & 31;
    const int wave   = threadIdx.x >> 5;
    const int m      = lane & 15;
    const int colOff = (lane < 16) ? 0 : 2;
    const int tileBase = (blockIdx.x * 8 + wave) * 16;
    const int k = tileBase + m;

    float lossC = 0.0f, sqC = 0.0f;

    if (tileBase < N) {                        // wave-uniform; EXEC stays full
        bool  valid = (k < N);
        float wv    = valid ? 1.0f : 0.0f;
        int   kc    = valid ? k : (N - 1);
        int   b     = list[baseSelf + kc];
        const float* uRow   = uw + (size_t)ub[b] * DIM;
        const float* negRow = iw + (size_t)jbv[b] * DIM;
        int   bl = list[baseOpp + (kc % mod)];
        int   il = ibv[bl];
        const float* pI = iw + (size_t)il * DIM;
        const float* pN = nw + (size_t)il * DIM;

        v8f cP = {};
        v8f cN = {};
        float sq = 0.0f;
#pragma unroll
        for (int c = 0; c < 16; ++c) {
            int col = 4 * c + colOff;
            float2 u2 = ld2(uRow + col);
            float2 n2 = ld2(negRow + col);
            float2 pi = ld2(pI + col);
            float2 pn = ld2(pN + col);
            float2 lk;
            lk.x = clip1(pi.x) + pn.x;
            lk.y = clip1(pi.y) + pn.y;
            sq += wv * (lk.x * lk.x + lk.y * lk.y);
            v2f av; av[0] = u2.x; av[1] = u2.y;
            v2f bp; bp[0] = lk.x; bp[1] = lk.y;
            v2f bn; bn[0] = n2.x; bn[1] = n2.y;
            cP = __builtin_amdgcn_wmma_f32_16x16x4_f32(false, av, false, bp,
                                                       (short)0, cP, false, false);
            cN = __builtin_amdgcn_wmma_f32_16x16x4_f32(false, av, false, bn,
                                                       (short)0, cN, false, false);
        }
        float pPos = wmmaDiag(cP, m);
        float pNeg = wmmaDiag(cN, m);
        if (lane < 16 && valid) lossC = softplus_(pNeg - pPos);
        sqC = sq;
    }

    lossC = waveSum(lossC);
    sqC   = waveSum(sqC);

    __shared__ float sl[8], ss[8];
    if (lane == 0) { sl[wave] = lossC; ss[wave] = sqC; }
    __syncthreads();
    if (threadIdx.x == 0) {
        float L = 0.0f, S = 0.0f;
#pragma unroll
        for (int i = 0; i < 8; ++i) { L += sl[i]; S += ss[i]; }
        float* dst = p3 + ((size_t)g * GRID3 + blockIdx.x) * 2;
        dst[0] = L;
        dst[1] = S;
    }
}

// --------------------------- finalize --------------------------------------

__device__ __forceinline__ float blockSum256(float v, float* red) {
    int t = threadIdx.x;
    red[t] = v;
    __syncthreads();
    for (int o = 128; o > 0; o >>= 1) {
        if (t < o) red[t] += red[t + o];
        __syncthreads();
    }
    float r = red[0];
    __syncthreads();
    return r;
}

__global__ void __launch_bounds__(256) k_final(const float* __restrict__ p2,
                                               const float* __restrict__ p3,
                                               const int* __restrict__ meta,
                                               float* __restrict__ out) {
    __shared__ float red[256];
    int t = threadIdx.x;
    float a0 = 0, a1 = 0, a2 = 0, a3 = 0, a4 = 0, a5 = 0;
    for (int i = t; i < GRID2; i += 256) {
        a0 += p2[(size_t)i * 2 + 0];
        a1 += p2[(size_t)i * 2 + 1];
    }
    for (int i = t; i < GRID3; i += 256) {
        a2 += p3[(size_t)i * 2 + 0];                    // male loss
        a3 += p3[(size_t)i * 2 + 1];                    // male like^2
        a4 += p3[((size_t)GRID3 + i) * 2 + 0];          // female loss
        a5 += p3[((size_t)GRID3 + i) * 2 + 1];          // female like^2
    }
    a0 = blockSum256(a0, red);
    a1 = blockSum256(a1, red);
    a2 = blockSum256(a2, red);
    a3 = blockSum256(a3, red);
    a4 = blockSum256(a4, red);
    a5 = blockSum256(a5, red);

    if (t == 0) {
        float fNm = (float)((meta[0] > 0) ? meta[0] : 1);
        float fNf = (float)((meta[1] > 0) ? meta[1] : 1);
        float invB = 1.0f / (float)B_SZ;
        float lossAdd = a0 * invB;
        float l2 = 0.01f * a1 * invB;
        float lm = a2 / fNm;
        float lf = a4 / fNf;
        float l22 = 0.01f * (a3 / fNm) + 0.01f * (a5 / fNf);
        out[0] = lossAdd + l2;     // loss_task
        out[1] = l2;               // l2_regulization
        out[2] = lm + lf + l22;    // loss_add_item
    }
}

// --------------------------- launcher --------------------------------------

extern "C" void kernel_launch(void* const* d_in, const int* in_sizes, int n_in,
                              void* d_out, int out_size, void* d_ws, size_t ws_size,
                              hipStream_t stream) {
    (void)in_sizes; (void)n_in; (void)out_size; (void)ws_size;

    const int*   ub  = (const int*)d_in[0];   // u_batch  (B)
    const int*   ibv = (const int*)d_in[1];   // i_batch  (B)
    const int*   jbv = (const int*)d_in[2];   // j_batch  (B)
    const int*   uf  = (const int*)d_in[3];   // users_features (U)
    const float* uw  = (const float*)d_in[4]; // user_w  (U x 64)
    const float* iw  = (const float*)d_in[5]; // item_w  (I x 64)
    const float* nw  = (const float*)d_in[6]; // noise_w (I x 64)

    char* ws = (char*)d_ws;
    int*   meta   = (int*)(ws + 0);                         // Nm, Nf         (64 B)
    int*   blkCnt = (int*)(ws + 64);                        // 1024 ints      (4 KB)
    int*   blkOff = (int*)(ws + 64 + 4096);                 // 1024 ints      (4 KB)
    int*   list   = (int*)(ws + 64 + 8192);                 // B ints         (1 MB)
    float* p2     = (float*)(ws + 8256 + (size_t)B_SZ * 4); // 2048*2 floats  (16 KB)
    float* p3     = p2 + (size_t)GRID2 * 2;                 // 2*2048*2       (32 KB)
    float* out    = (float*)d_out;

    k_count  <<<CBLOCKS, 256, 0, stream>>>(ub, uf, blkCnt);
    k_scan   <<<1, 1024, 0, stream>>>(blkCnt, blkOff, meta);
    k_scatter<<<CBLOCKS, 256, 0, stream>>>(ub, uf, blkOff, meta, list);
    k_main   <<<GRID2, 256, 0, stream>>>(ub, ibv, jbv, uw, iw, nw, p2);
    k_fake   <<<dim3(GRID3, 2), 256, 0, stream>>>(ub, ibv, jbv, uw, iw, nw, meta, list, p3);
    k_final  <<<1, 256, 0, stream>>>(p2, p3, meta, out);
}